// LossV2_18940805776109
// MI455X (gfx1250) — compile-verified
//
#include <hip/hip_runtime.h>
#include <math.h>

// Problem constants (fixed by the reference file)
#define NVAR    10000
#define NCLAUSE 40000
#define NNODE   (NVAR + NCLAUSE)
#define KLIT    3

#define BLK 256

__device__ __forceinline__ float softplus_f(float x) {
    // numerically stable: softplus(x) = max(x,0) + log1p(exp(-|x|))
    return fmaxf(x, 0.0f) + log1pf(__expf(-fabsf(x)));
}

// wave32 + LDS block reduction; result valid in thread 0
__device__ __forceinline__ float block_reduce(float v, float* swarp) {
    #pragma unroll
    for (int off = 16; off > 0; off >>= 1)
        v += __shfl_xor(v, off, 32);
    const int lane = threadIdx.x & 31;
    const int wid  = threadIdx.x >> 5;
    if (lane == 0) swarp[wid] = v;
    __syncthreads();
    if (wid == 0) {
        v = (lane < (int)(blockDim.x >> 5)) ? swarp[lane] : 0.0f;
        #pragma unroll
        for (int off = 16; off > 0; off >>= 1)
            v += __shfl_xor(v, off, 32);
    }
    return v;
}

__global__ void init_flags(uint4* __restrict__ flags128, int n128) {
    int i = blockIdx.x * BLK + threadIdx.x;
    if (i < n128) flags128[i] = make_uint4(0u, 0u, 0u, 0u);
}

// Pass 1: clause positions, 4 elements (one float4) per lane per iteration.
// Streams out/y with B128 loads, async-stages the 123 MB clause-index stream
// through LDS with CDNA5 async global->LDS DMA (3x B128 DMA per lane -> 12 KB
// tile), byte-scatters "active" flags (race-free: all writers store 1).
__global__ void clause_pass(const float4* __restrict__ out4,
                            const float4* __restrict__ y4,
                            const int*    __restrict__ clauses,
                            unsigned char* __restrict__ flags,
                            float* __restrict__ partials,
                            int total4, int iters, float scale1) {
    __shared__ alignas(16) int smem[BLK * 4 * KLIT];   // 12 KiB staging tile
    __shared__ float swarp[BLK / 32];
    const int t = threadIdx.x;
    const unsigned ldsBase = (unsigned)(size_t)(&smem[0]); // LDS byte offset
    const int stride = (int)gridDim.x * BLK;               // in float4 chunks
    float acc = 0.0f;

    for (int it = 0; it < iters; ++it) {
        const int blockStart = (int)blockIdx.x * BLK + it * stride; // chunk idx
        const int rem   = total4 - blockStart;
        const int bytes = (rem < BLK ? rem : BLK) * (4 * KLIT * 4); // <=12288

        // --- async DMA: stage this block's 1024*3 int32 indices into LDS ---
        // GVS mode: saddr(SGPR pair) + per-lane 32-bit offset; VDST = LDS addr.
        const char* gsrc = (const char*)clauses + (size_t)blockStart * (4 * KLIT * 4);
        #pragma unroll
        for (int c = 0; c < 3; ++c) {
            const unsigned off = (unsigned)(t * 16 + c * 4096);
            if (off < (unsigned)bytes) {
                asm volatile("global_load_async_to_lds_b128 %0, %1, %2"
                             :: "v"(ldsBase + off), "v"(off), "s"(gsrc) : "memory");
            }
        }

        // overlap: softplus/BCE math on out/y while the DMA runs
        const int j = blockStart + t;                 // float4 chunk index
        float4 x = make_float4(0.f, 0.f, 0.f, 0.f);
        bool inb = (j < total4);
        int  base = 0;
        if (inb) {
            const int b    = (4 * j) / NCLAUSE;       // chunk never crosses b
            const int addr = j + (b + 1) * (NVAR / 4);// == (b*NNODE+NVAR+c)/4
            x = out4[addr];
            const float4 ty = y4[addr];
            acc += (softplus_f(x.x) - x.x * ty.x) + (softplus_f(x.y) - x.y * ty.y)
                 + (softplus_f(x.z) - x.z * ty.z) + (softplus_f(x.w) - x.w * ty.w);
            base = b * NVAR;
            if (it + 1 < iters)                       // keep streams flowing
                __builtin_prefetch((const void*)(out4 + addr + stride), 0, 3);
        }

        asm volatile("s_wait_asynccnt 0" ::: "memory");
        __syncthreads();                              // tile visible to all waves

        if (inb) {
            const int4* sm4 = (const int4*)smem;      // 3x ds_load_b128 per lane
            const int4 c0 = sm4[t * 3 + 0];
            const int4 c1 = sm4[t * 3 + 1];
            const int4 c2 = sm4[t * 3 + 2];
            if (x.x > 0.0f) { flags[base + c0.x] = 1; flags[base + c0.y] = 1; flags[base + c0.z] = 1; }
            if (x.y > 0.0f) { flags[base + c0.w] = 1; flags[base + c1.x] = 1; flags[base + c1.y] = 1; }
            if (x.z > 0.0f) { flags[base + c1.z] = 1; flags[base + c1.w] = 1; flags[base + c2.x] = 1; }
            if (x.w > 0.0f) { flags[base + c2.y] = 1; flags[base + c2.z] = 1; flags[base + c2.w] = 1; }
        }
        __syncthreads();                              // tile dead before next DMA
    }

    const float tot = block_reduce(acc, swarp);
    if (t == 0) partials[blockIdx.x] = tot * scale1;
}

// Pass 2: var positions; flags (2.56 MB << 192 MB L2) read as uchar4.
__global__ void var_pass(const float4* __restrict__ out4,
                         const uchar4* __restrict__ flags4,
                         float* __restrict__ partials,
                         int total4, int iters, float scale2) {
    __shared__ float swarp[BLK / 32];
    const int t = threadIdx.x;
    const int stride = (int)gridDim.x * BLK;          // in float4 chunks
    float acc = 0.0f;
    for (int it = 0; it < iters; ++it) {
        const int j = (int)blockIdx.x * BLK + it * stride + t;
        if (j < total4) {
            const int b    = (4 * j) / NVAR;          // chunk never crosses b
            const int addr = j + b * (NCLAUSE / 4);   // == (b*NNODE + v)/4
            const float4 x = out4[addr];
            const uchar4 f = flags4[j];
            acc += (softplus_f(x.x) - (f.x ? x.x : 0.0f))
                 + (softplus_f(x.y) - (f.y ? x.y : 0.0f))
                 + (softplus_f(x.z) - (f.z ? x.z : 0.0f))
                 + (softplus_f(x.w) - (f.w ? x.w : 0.0f));
            if (it + 1 < iters)
                __builtin_prefetch((const void*)(out4 + addr + stride), 0, 3);
        }
    }
    const float tot = block_reduce(acc, swarp);
    if (t == 0) partials[blockIdx.x] = tot * scale2;
}

// Deterministic final sum (fixed accumulation order, no float atomics).
__global__ void final_reduce(const float* __restrict__ partials, int n,
                             float constTerm, float* __restrict__ result) {
    __shared__ float swarp[BLK / 32];
    float acc = 0.0f;
    for (int i = threadIdx.x; i < n; i += BLK) acc += partials[i];
    const float tot = block_reduce(acc, swarp);
    if (threadIdx.x == 0) result[0] = constTerm + tot;
}

extern "C" void kernel_launch(void* const* d_in, const int* in_sizes, int n_in,
                              void* d_out, int out_size, void* d_ws, size_t ws_size,
                              hipStream_t stream) {
    // dict order: out, y, mask, clauses, n_var, n_clause
    const float4* out4    = (const float4*)d_in[0];
    const float4* y4      = (const float4*)d_in[1];
    const int*    clauses = (const int*)d_in[3];      // mask (d_in[2]) is implicit
    float*        result  = (float*)d_out;

    const int B = in_sizes[0] / NNODE;
    const int clauseTotal = B * NCLAUSE;              // 10,240,000
    const int varTotal    = B * NVAR;                 //  2,560,000
    const int clauseT4    = clauseTotal / 4;          //  2,560,000 chunks
    const int varT4       = varTotal / 4;             //    640,000 chunks

    // workspace: [flags: varTotal bytes, 16B-rounded][partials floats]
    unsigned char* flags = (unsigned char*)d_ws;
    const size_t flagsBytes = ((size_t)varTotal + 15) & ~(size_t)15;
    float* partials1 = (float*)((char*)d_ws + flagsBytes);

    const int ITERS_C = 5;                            // 2000 blocks * 256 * 5 chunks
    const int ITERS_V = 4;                            //  625 blocks * 256 * 4 chunks
    const int nb1 = (clauseT4 + BLK * ITERS_C - 1) / (BLK * ITERS_C);
    const int nb2 = (varT4    + BLK * ITERS_V - 1) / (BLK * ITERS_V);
    float* partials2 = partials1 + nb1;

    const double ln2    = 0.69314718055994530942;
    const double scale1 = 1.0 / ((double)NNODE * (double)NCLAUSE);
    const double scale2 = 2.0 / (double)NNODE;
    const double constTerm = (double)B * (double)NVAR    * ln2 * scale1
                           + (double)B * (double)NCLAUSE * ln2 * scale2;

    const int n128 = (int)(flagsBytes / 16);          // zero flags as uint4
    init_flags<<<(n128 + BLK - 1) / BLK, BLK, 0, stream>>>((uint4*)d_ws, n128);
    clause_pass<<<nb1, BLK, 0, stream>>>(out4, y4, clauses, flags, partials1,
                                         clauseT4, ITERS_C, (float)scale1);
    var_pass<<<nb2, BLK, 0, stream>>>(out4, (const uchar4*)flags, partials2,
                                      varT4, ITERS_V, (float)scale2);
    final_reduce<<<1, BLK, 0, stream>>>(partials1, nb1 + nb2,
                                        (float)constTerm, result);
}